// AttrSoftLoss_55602646614713
// MI455X (gfx1250) — compile-verified
//
#include <hip/hip_runtime.h>
#include <hip/hip_bf16.h>

#define B_ROWS 8192
#define C_COLS 1024
#define MASK_FRAC 0.95f

typedef __attribute__((ext_vector_type(4))) unsigned int uint32x4;
typedef __attribute__((ext_vector_type(4))) int int32x4;
typedef __attribute__((ext_vector_type(8))) int int32x8;
typedef __attribute__((ext_vector_type(2))) float v2f;
typedef __attribute__((ext_vector_type(8))) float v8f;

__device__ __forceinline__ unsigned rotl32(unsigned x, unsigned r) {
  return (x << r) | (x >> (32u - r));
}

// JAX Threefry-2x32 (20 rounds), key=(k0,k1), counter=(x0,x1)
__device__ __forceinline__ void threefry2x32(unsigned k0, unsigned k1,
                                             unsigned x0, unsigned x1,
                                             unsigned& o0, unsigned& o1) {
  const unsigned ks2 = k0 ^ k1 ^ 0x1BD11BDAu;
  x0 += k0; x1 += k1;
#define TF_R(r) { x0 += x1; x1 = rotl32(x1, r); x1 ^= x0; }
  TF_R(13u) TF_R(15u) TF_R(26u) TF_R(6u)
  x0 += k1;  x1 += ks2 + 1u;
  TF_R(17u) TF_R(29u) TF_R(16u) TF_R(24u)
  x0 += ks2; x1 += k0 + 2u;
  TF_R(13u) TF_R(15u) TF_R(26u) TF_R(6u)
  x0 += k0;  x1 += k1 + 3u;
  TF_R(17u) TF_R(29u) TF_R(16u) TF_R(24u)
  x0 += k1;  x1 += ks2 + 4u;
  TF_R(13u) TF_R(15u) TF_R(26u) TF_R(6u)
  x0 += ks2; x1 += k0 + 5u;
#undef TF_R
  o0 = x0; o1 = x1;
}

// bits[g] of jax.random.uniform(key(42), (B,C)) before mantissa mapping.
// JAX splits the flat iota in half: (x0,x1)=(g, g+N/2) -> out0 for g<N/2,
// (x0,x1)=(g-N/2, g) -> out1 otherwise. impl key of jax.random.key(42) = (0,42).
__device__ __forceinline__ unsigned jax_bits_at(unsigned long long g) {
  const unsigned long long HALF = (unsigned long long)B_ROWS * C_COLS / 2ull;
  unsigned o0, o1;
  if (g < HALF) {
    threefry2x32(0u, 42u, (unsigned)g, (unsigned)(g + HALF), o0, o1);
    return o0;
  }
  threefry2x32(0u, 42u, (unsigned)(g - HALF), (unsigned)g, o0, o1);
  return o1;
}

__device__ __forceinline__ float log_sigmoid(float x) {
  return fminf(x, 0.0f) - log1pf(expf(-fabsf(x)));
}

// ---- TDM: DMA one contiguous row (C_COLS 4-byte elements) into LDS ----
// D# layout per CDNA5 ISA.
__device__ __forceinline__ void tdm_load_row(unsigned lds_off, const void* gptr) {
  const unsigned long long ga = (unsigned long long)gptr;
  uint32x4 g0;
  g0.x = 1u;                                            // count=1, user D#
  g0.y = lds_off;                                       // lds_addr (bytes)
  g0.z = (unsigned)(ga & 0xFFFFFFFFull);                // global_addr[31:0]
  g0.w = (unsigned)((ga >> 32) & 0x01FFFFFFull)         // global_addr[56:32]
         | 0x80000000u;                                 // type=2 ("image")
  int32x8 g1;
  g1[0] = 0x00020000;                                   // data_size=2 (4 bytes)
  g1[1] = (int)(((unsigned)C_COLS & 0xFFFFu) << 16);    // tensor_dim0[15:0]
  g1[2] = (int)(((unsigned)B_ROWS & 0xFFFFu) << 16);    // dim0 hi=0 | tensor_dim1 lo
  g1[3] = (int)(((unsigned)C_COLS & 0xFFFFu) << 16);    // dim1 hi=0 | tile_dim0=1024
  g1[4] = 1;                                            // tile_dim1=1, tile_dim2=0
  g1[5] = C_COLS;                                       // tensor_dim0_stride[31:0]
  g1[6] = (int)(((unsigned)C_COLS & 0xFFFFu) << 16);    // s0 hi=0 | tensor_dim1_stride lo
  g1[7] = 0;                                            // tensor_dim1_stride hi
  const int32x4 z4 = {0, 0, 0, 0};
  const int32x8 z8 = {0, 0, 0, 0, 0, 0, 0, 0};
  // 6-arg form on this toolchain: (g0, g1, g2, g3, g4, cpol)
  __builtin_amdgcn_tensor_load_to_lds(g0, g1, z4, z4, z8, 0);
}

// One 256-thread block per row. Exact JAX-equivalent masking via 3-pass
// radix-select on 24-bit keys + stable tie-break; writes row sum of masked
// log-sigmoid terms to row_sums[row].
__global__ __launch_bounds__(256) void attr_loss_rows(
    const float* __restrict__ scores, const int* __restrict__ attrs,
    float* __restrict__ row_sums) {
  __shared__ float    s_scores[C_COLS];
  __shared__ int      s_attrs[C_COLS];
  __shared__ unsigned s_hist[256];
  __shared__ unsigned s_scan[256];
  __shared__ float    s_red[256];
  __shared__ unsigned s_bin, s_rem;

  const int tid = threadIdx.x;
  const int row = blockIdx.x;

  if (tid < 32) {  // wave 0 issues both TDM transfers and waits on them
    tdm_load_row((unsigned)(unsigned long long)&s_scores[0],
                 scores + (size_t)row * C_COLS);
    tdm_load_row((unsigned)(unsigned long long)&s_attrs[0],
                 attrs + (size_t)row * C_COLS);
    __builtin_amdgcn_s_wait_tensorcnt(0);
  }
  __syncthreads();

  // 4 contiguous columns per thread, all state in registers
  float sc[4]; int at[4]; unsigned key[4];
  int zc = 0;
  const int c0 = tid * 4;
#pragma unroll
  for (int j = 0; j < 4; ++j) {
    sc[j] = s_scores[c0 + j];
    at[j] = s_attrs[c0 + j];
    const unsigned long long g = (unsigned long long)row * C_COLS + (c0 + j);
    const unsigned bits = jax_bits_at(g);
    const bool is_zero = (at[j] == 0);
    // u = bitcast((bits>>9)|0x3f800000) - 1  is monotone in (bits>>9);
    // non-zero labels get u=2.0 -> sentinel above any real 23-bit key.
    key[j] = is_zero ? (bits >> 9) : 0x00FFFFFFu;
    zc += is_zero ? 1 : 0;
  }

  // n_zero reduction
  s_scan[tid] = (unsigned)zc;
  __syncthreads();
  for (int s = 128; s > 0; s >>= 1) {
    if (tid < s) s_scan[tid] += s_scan[tid + s];
    __syncthreads();
  }
  const int n_zero = (int)s_scan[0];
  __syncthreads();
  const int kk = (int)rintf((float)n_zero * MASK_FRAC);  // jnp.round (RNE)

  unsigned thr = 0xFFFFFFFFu, t_need = 0;
  const bool do_mask = (kk > 0);
  if (do_mask) {  // uniform per block
    unsigned prefix = 0, remaining = (unsigned)kk;
    for (int pass = 0; pass < 3; ++pass) {
      const int shift = 16 - 8 * pass;
      s_hist[tid] = 0;
      __syncthreads();
#pragma unroll
      for (int j = 0; j < 4; ++j) {
        if ((key[j] >> (shift + 8)) == prefix)
          atomicAdd(&s_hist[(key[j] >> shift) & 0xFFu], 1u);
      }
      __syncthreads();
      if (tid == 0) {
        unsigned cum = 0, b = 0;
        for (; b < 255u; ++b) {
          const unsigned h = s_hist[b];
          if (cum + h >= remaining) break;
          cum += h;
        }
        s_bin = b; s_rem = remaining - cum;
      }
      __syncthreads();
      prefix = (prefix << 8) | s_bin;
      remaining = s_rem;
      __syncthreads();
    }
    thr = prefix;       // k-th smallest key value
    t_need = remaining; // how many ties (by column order) fall below rank k
  }

  // stable tie ordering: exclusive count of (key==thr) before each element
  unsigned tloc[4]; unsigned tcnt = 0;
#pragma unroll
  for (int j = 0; j < 4; ++j) { tloc[j] = tcnt; tcnt += (key[j] == thr) ? 1u : 0u; }
  s_scan[tid] = tcnt;
  __syncthreads();
  for (int off = 1; off < 256; off <<= 1) {
    const unsigned v = (tid >= off) ? s_scan[tid - off] : 0u;
    __syncthreads();
    s_scan[tid] += v;
    __syncthreads();
  }
  const unsigned texcl = (tid == 0) ? 0u : s_scan[tid - 1];

  float acc = 0.0f;
#pragma unroll
  for (int j = 0; j < 4; ++j) {
    const bool masked =
        do_mask && ((key[j] < thr) ||
                    ((key[j] == thr) && (texcl + tloc[j] < t_need)));
    const float v = at[j] ? log_sigmoid(sc[j]) : log_sigmoid(-sc[j]);
    acc += masked ? 0.0f : v;
  }
  s_red[tid] = acc;
  __syncthreads();
  for (int s = 128; s > 0; s >>= 1) {
    if (tid < s) s_red[tid] += s_red[tid + s];
    __syncthreads();
  }
  if (tid == 0) row_sums[row] = s_red[0];
}

// Single-wave final reduction of 8192 row sums using V_WMMA_F32_16X16X4_F32:
// each 64-float chunk is a 16x4 A tile; B = ones(4x16); C accumulates row sums.
__global__ __launch_bounds__(32) void wmma_reduce(const float* __restrict__ ws,
                                                  float* __restrict__ out) {
  const int lane = threadIdx.x;          // 0..31, EXEC all ones
  const int m = lane & 15;
  const int kb = (lane & 16) ? 2 : 0;    // A layout: lanes 0-15 hold K=0,1; 16-31 hold K=2,3
  v2f b; b.x = 1.0f; b.y = 1.0f;
  v8f c = {};
  for (int t = 0; t < B_ROWS; t += 64) {
    v2f a;
    a.x = ws[t + m * 4 + kb];
    a.y = ws[t + m * 4 + kb + 1];
    c = __builtin_amdgcn_wmma_f32_16x16x4_f32(
        /*neg_a=*/false, a, /*neg_b=*/false, b,
        /*c_mod=*/(short)0, c, /*reuse_a=*/false, /*reuse_b=*/false);
  }
  // D[m][n] = rowsum_m for every n. Lane 0 holds M=0..7 across its 8 VGPRs,
  // lane 16 holds M=8..15.
  float loc = c[0] + c[1] + c[2] + c[3] + c[4] + c[5] + c[6] + c[7];
  const float t0  = __shfl(loc, 0, 32);
  const float t16 = __shfl(loc, 16, 32);
  if (lane == 0)
    out[0] = -(t0 + t16) / ((float)B_ROWS * (float)C_COLS);
}

extern "C" void kernel_launch(void* const* d_in, const int* in_sizes, int n_in,
                              void* d_out, int out_size, void* d_ws, size_t ws_size,
                              hipStream_t stream) {
  (void)in_sizes; (void)n_in; (void)out_size; (void)ws_size;
  const float* scores = (const float*)d_in[0];
  const int* attrs    = (const int*)d_in[1];
  float* out = (float*)d_out;
  float* ws  = (float*)d_ws;  // 8192 floats of per-row sums

  attr_loss_rows<<<B_ROWS, 256, 0, stream>>>(scores, attrs, ws);
  wmma_reduce<<<1, 32, 0, stream>>>(ws, out);
}